// GATConv_16260746183220
// MI455X (gfx1250) — compile-verified
//
#include <hip/hip_runtime.h>

// ---------------------------------------------------------------------------
// GATConv for MI455X (gfx1250, wave32).  B=8, L=256, E=256, H=8, A=32.
// fp32 end-to-end; GEMMs via V_WMMA_F32_16X16X4_F32; panel staging via
// GLOBAL_LOAD_ASYNC_TO_LDS_B128 (ASYNCcnt path).
//
// Roofline: ~1.1 GFLOP total, ~11 MB unique traffic (all L2-resident at
// 192 MB).  Dominant cost is the non-separable 134M-element |k_i + q_j|
// reduction -> minimize VALU ops/element (2: add + fma-with-abs-modifier)
// and LDS ops/element (0.25: ds_load_b128 per 4 elements).
// ---------------------------------------------------------------------------

typedef __attribute__((ext_vector_type(2))) float v2f;
typedef __attribute__((ext_vector_type(8))) float v8f;
typedef __attribute__((ext_vector_type(4))) int   v4i;

#define B_    8
#define L_    256
#define E_    256
#define H_    8
#define A_    32
#define TWOE_ 512
#define NEG_MIN (-3.402823466e38f)   // finfo(float32).min

// ---- CDNA5 async global->LDS staging (ASYNCcnt path), with safe fallback ---
#if defined(__has_builtin)
#  if __has_builtin(__builtin_amdgcn_global_load_async_to_lds_b128)
#    define HAVE_ASYNC 1
#  endif
#endif
#ifndef HAVE_ASYNC
#  define HAVE_ASYNC 0
#endif

#if HAVE_ASYNC
// signature (from hipcc diagnostics): (int4 AS1* src, int4 AS3* lds, Ii, Ii)
#  define GLD_ASYNC_B128(gp, lp)                                        \
     __builtin_amdgcn_global_load_async_to_lds_b128(                    \
         (__attribute__((address_space(1))) v4i*)(gp),                  \
         (__attribute__((address_space(3))) v4i*)(lp), 0, 0)
#  if __has_builtin(__builtin_amdgcn_s_wait_asynccnt)
#    define WAIT_ASYNC() __builtin_amdgcn_s_wait_asynccnt(0)
#  else
#    define WAIT_ASYNC() asm volatile("s_wait_asynccnt 0" ::: "memory")
#  endif
#endif

__device__ __forceinline__ v8f wmma_f32_16x16x4(v2f a, v2f b, v8f c) {
    // (neg_a, A, neg_b, B, c_mod, C, reuse_a, reuse_b)
    return __builtin_amdgcn_wmma_f32_16x16x4_f32(
        false, a, false, b, (short)0, c, false, false);
}

// ---------------------------------------------------------------------------
// Kernel 1: qk = X @ W_qk^T.  X:(2048,256), W_qk:(512,256) row-major.
// One wave per 16x64 macro-tile: A fragment reused across 4 N-tiles
// (1.25 loads per wmma instead of 2).  Scatter into (b,h,l,a) q/k buffers.
// ---------------------------------------------------------------------------
__global__ __launch_bounds__(32)
void qk_proj_kernel(const float* __restrict__ X,
                    const float* __restrict__ Wqk,
                    float* __restrict__ qbuf,
                    float* __restrict__ kbuf) {
    const int lane = threadIdx.x;
    const int half = lane >> 4;          // lane-half selects K pair {0,1}/{2,3}
    const int l16  = lane & 15;
    const int m0   = blockIdx.x * 16;    // row tile in B*L
    const int n0   = blockIdx.y * 64;    // 4 column tiles in 2E

    const float* arow = X   + (size_t)(m0 + l16) * E_;   // A: M = l16
    const float* brow = Wqk + (size_t)(n0 + l16) * E_;   // B: N = l16 (+16*nt)

    v8f acc[4] = {{}, {}, {}, {}};
    #pragma unroll 4
    for (int kk = 0; kk < E_; kk += 4) {
        const int o = kk + 2 * half;                     // K = o, o+1
        v2f a  = *(const v2f*)(arow + o);
        v2f b0 = *(const v2f*)(brow             + o);
        v2f b1 = *(const v2f*)(brow + 16 * E_   + o);
        v2f b2 = *(const v2f*)(brow + 32 * E_   + o);
        v2f b3 = *(const v2f*)(brow + 48 * E_   + o);
        acc[0] = wmma_f32_16x16x4(a, b0, acc[0]);
        acc[1] = wmma_f32_16x16x4(a, b1, acc[1]);
        acc[2] = wmma_f32_16x16x4(a, b2, acc[2]);
        acc[3] = wmma_f32_16x16x4(a, b3, acc[3]);
    }

    // D element (M = r + 8*half, N = l16); split columns into q / k halves.
    #pragma unroll
    for (int nt = 0; nt < 4; ++nt) {
        #pragma unroll
        for (int r = 0; r < 8; ++r) {
            const int m  = m0 + r + 8 * half;
            const int n  = n0 + nt * 16 + l16;
            const int bb = m >> 8;           // / L
            const int i  = m & (L_ - 1);
            const int h  = n >> 6;           // / (2A)
            const int c  = n & 63;
            const int base = (((bb * H_ + h) * L_) + i) * A_;
            const float v = acc[nt][r];
            if (c < A_) qbuf[base + c]        = v;
            else        kbuf[base + (c - A_)] = v;
        }
    }
}

// ---------------------------------------------------------------------------
// Kernel 2: fused scores + masked softmax + attn@k for one (b,h) head and a
// 16-row i-tile.  Reference semantics (axis trick: i indexes k, j indexes q):
//   scores[i,j] = sum_a w[a]*lrelu(k[i,a] + q[j,a])
//               = 0.6*kw[i] + 0.6*qw[j] + sum_a (0.4*w[a])*|k[i,a]+q[j,a]|
// The 0.6*kw[i] term is row-constant -> cancels exactly in softmax: dropped.
//   attn = softmax_j(mask(scores));  vals[i,a] = sum_j attn[i,j]*k[j,a]
// ---------------------------------------------------------------------------
__global__ __launch_bounds__(256)
void attn_kernel(const float* __restrict__ qbuf,
                 const float* __restrict__ kbuf,
                 const unsigned char* __restrict__ adj,
                 const float* __restrict__ w_attn,
                 float* __restrict__ vals) {
    __shared__ float ksf[L_ * A_];      // 32 KB  k panel (values + row vectors)
    __shared__ float qsf[L_ * A_];      // 32 KB  q panel (score columns)
    __shared__ float wsh[A_];
    __shared__ float qw6[L_];           // 1 KB   0.6 * (q[j] . w)
    __shared__ float sc[16 * L_];       // 16 KB  scores -> attn
    __shared__ float pbuf[16 * 16];
    __shared__ float rowmax[16];
    __shared__ float rowsum[16];
    __shared__ float pd[8 * 256];       // 8 KB   per-wave WMMA partials

    const int t  = threadIdx.x;
    const int bh = blockIdx.y;          // b*H + h
    const int b  = bh >> 3;
    const int h  = bh & (H_ - 1);
    const int i0 = blockIdx.x * 16;

    // ---- stage q,k head panels (256x32 f32 each) -------------------------
    const float4* ks4 = (const float4*)(kbuf + (size_t)bh * L_ * A_);
    const float4* qs4 = (const float4*)(qbuf + (size_t)bh * L_ * A_);
    float4* kd = (float4*)ksf;
    float4* qd = (float4*)qsf;
#if HAVE_ASYNC
    #pragma unroll
    for (int r = 0; r < 8; ++r) {
        GLD_ASYNC_B128(ks4 + t + r * 256, kd + t + r * 256);
        GLD_ASYNC_B128(qs4 + t + r * 256, qd + t + r * 256);
    }
    if (t < A_) wsh[t] = w_attn[t];
    WAIT_ASYNC();
#else
    #pragma unroll
    for (int r = 0; r < 8; ++r) {
        kd[t + r * 256] = ks4[t + r * 256];
        qd[t + r * 256] = qs4[t + r * 256];
    }
    if (t < A_) wsh[t] = w_attn[t];
#endif
    __syncthreads();

    // ---- precompute qw6[j] = 0.6 * sum_a w[a]*q[j,a]  (one j per thread) --
    {
        const float4* qr = (const float4*)(qsf + t * A_);
        float s = 0.f;
        #pragma unroll
        for (int p = 0; p < 8; ++p) {
            const float4 qv = qr[p];
            s += wsh[4 * p + 0] * qv.x + wsh[4 * p + 1] * qv.y
               + wsh[4 * p + 2] * qv.z + wsh[4 * p + 3] * qv.w;
        }
        qw6[t] = 0.6f * s;
    }

    // register-cache this row's k vector and the 0.4-scaled weights
    const int il = t & 15;
    const int jg = t >> 4;
    float kr[A_], wr[A_];
    #pragma unroll
    for (int a = 0; a < A_; ++a) {
        kr[a] = ksf[(i0 + il) * A_ + a];
        wr[a] = 0.4f * wsh[a];
    }
    __syncthreads();                    // qw6 ready

    // ---- scores: row i = t&15, columns j = jg*16..+15 ---------------------
    const unsigned char* adjrow = adj + ((size_t)(b * L_ + i0 + il)) * L_;
    const uint4 am = *(const uint4*)(adjrow + jg * 16);   // 16 mask bytes
    const unsigned aw[4] = {am.x, am.y, am.z, am.w};

    float my[16];
    #pragma unroll
    for (int jj = 0; jj < 16; ++jj) {
        const int j = jg * 16 + jj;
        const float4* qr = (const float4*)(qsf + j * A_);
        float a0 = 0.f, a1 = 0.f;
        #pragma unroll
        for (int p = 0; p < 8; ++p) {
            const float4 qv = qr[p];                      // ds_load_b128
            a0 = fmaf(wr[4 * p + 0], fabsf(kr[4 * p + 0] + qv.x), a0);
            a1 = fmaf(wr[4 * p + 1], fabsf(kr[4 * p + 1] + qv.y), a1);
            a0 = fmaf(wr[4 * p + 2], fabsf(kr[4 * p + 2] + qv.z), a0);
            a1 = fmaf(wr[4 * p + 3], fabsf(kr[4 * p + 3] + qv.w), a1);
        }
        const unsigned ab = (aw[jj >> 2] >> ((jj & 3) * 8)) & 0xffu;
        my[jj] = ab ? (qw6[j] + a0 + a1) : NEG_MIN;
    }

    // ---- masked softmax over j (row length 256, 16 threads per row) -------
    float pmax = my[0];
    #pragma unroll
    for (int jj = 1; jj < 16; ++jj) pmax = fmaxf(pmax, my[jj]);
    pbuf[il * 16 + jg] = pmax;
    __syncthreads();
    if (t < 16) {
        float mm = pbuf[t * 16];
        #pragma unroll
        for (int g = 1; g < 16; ++g) mm = fmaxf(mm, pbuf[t * 16 + g]);
        rowmax[t] = mm;
    }
    __syncthreads();
    const float rm = rowmax[il];
    float psum = 0.f;
    #pragma unroll
    for (int jj = 0; jj < 16; ++jj) {
        const float e = __expf(my[jj] - rm);   // masked -> exp(-inf) = 0
        my[jj] = e;
        psum += e;
    }
    pbuf[il * 16 + jg] = psum;
    __syncthreads();
    if (t < 16) {
        float ss = 0.f;
        #pragma unroll
        for (int g = 0; g < 16; ++g) ss += pbuf[t * 16 + g];
        rowsum[t] = ss;
    }
    __syncthreads();
    const float inv = 1.f / rowsum[il];        // diag of adj guarantees sum>0
    #pragma unroll
    for (int jj = 0; jj < 16; ++jj)
        sc[il * L_ + jg * 16 + jj] = my[jj] * inv;
    __syncthreads();

    // ---- vals(16x32) = attn(16x256) @ k(256x32) via WMMA ------------------
    // 8 waves: wave w -> N-half (w&1), K-quarter (w>>1); 16 wmma each.
    {
        const int wvid = t >> 5;
        const int lane = t & 31;
        const int half = lane >> 4;
        const int l16  = lane & 15;
        const int nh   = wvid & 1;
        const int kq   = wvid >> 1;

        v8f acc = {};
        const float* arow = sc  + l16 * L_;           // A: M = l16 (attn row)
        const float* bcol = ksf + nh * 16 + l16;      // B: N = nh*16 + l16
        #pragma unroll 4
        for (int kk = kq * 64; kk < kq * 64 + 64; kk += 4) {
            const int k0 = kk + 2 * half;
            v2f a = *(const v2f*)(arow + k0);
            v2f bb;
            bb.x = bcol[(k0    ) * A_];
            bb.y = bcol[(k0 + 1) * A_];
            acc = wmma_f32_16x16x4(a, bb, acc);
        }
        #pragma unroll
        for (int r = 0; r < 8; ++r)
            pd[wvid * 256 + r * 32 + lane] = acc[r];
    }
    __syncthreads();

    // ---- reduce the 4 K-quarter partials, write vals (b, i, h*A + a) ------
    #pragma unroll
    for (int rep = 0; rep < 2; ++rep) {
        const int e   = t + rep * 256;     // 0..511 = (nh, r, lane)
        const int nh2 = e >> 8;
        const int r   = (e >> 5) & 7;
        const int ln  = e & 31;
        const int off = nh2 * 256 + r * 32 + ln;
        const float v = pd[off] + pd[512 + off] + pd[1024 + off] + pd[1536 + off];
        const int M     = r + 8 * (ln >> 4);
        const int a_out = nh2 * 16 + (ln & 15);
        vals[((size_t)(b * L_ + i0 + M)) * E_ + h * A_ + a_out] = v;
    }
}

// ---------------------------------------------------------------------------
// Kernel 3: out = vals @ W_out^T.  (2048,256)@(256,256) -> (2048,256).
// Same 16x64 macro-tile structure as kernel 1.
// ---------------------------------------------------------------------------
__global__ __launch_bounds__(32)
void out_proj_kernel(const float* __restrict__ V,
                     const float* __restrict__ Wout,
                     float* __restrict__ out) {
    const int lane = threadIdx.x;
    const int half = lane >> 4;
    const int l16  = lane & 15;
    const int m0   = blockIdx.x * 16;
    const int n0   = blockIdx.y * 64;

    const float* arow = V    + (size_t)(m0 + l16) * E_;
    const float* brow = Wout + (size_t)(n0 + l16) * E_;

    v8f acc[4] = {{}, {}, {}, {}};
    #pragma unroll 4
    for (int kk = 0; kk < E_; kk += 4) {
        const int o = kk + 2 * half;
        v2f a  = *(const v2f*)(arow + o);
        v2f b0 = *(const v2f*)(brow             + o);
        v2f b1 = *(const v2f*)(brow + 16 * E_   + o);
        v2f b2 = *(const v2f*)(brow + 32 * E_   + o);
        v2f b3 = *(const v2f*)(brow + 48 * E_   + o);
        acc[0] = wmma_f32_16x16x4(a, b0, acc[0]);
        acc[1] = wmma_f32_16x16x4(a, b1, acc[1]);
        acc[2] = wmma_f32_16x16x4(a, b2, acc[2]);
        acc[3] = wmma_f32_16x16x4(a, b3, acc[3]);
    }
    #pragma unroll
    for (int nt = 0; nt < 4; ++nt) {
        #pragma unroll
        for (int r = 0; r < 8; ++r) {
            const int m = m0 + r + 8 * half;
            out[(size_t)m * E_ + n0 + nt * 16 + l16] = acc[nt][r];
        }
    }
}

// ---------------------------------------------------------------------------
// Launch.  Inputs (setup_inputs order):
//   0 embeddings (B,L,E) f32 | 1 adj (B,L,L) bool(1B) | 2 W_qk (2E,E) f32
//   3 w_attn (A,) f32        | 4 W_out (E,E) f32
// Workspace: q (2MB) | k (2MB) | vals (2MB) -- all fully overwritten.
// ---------------------------------------------------------------------------
extern "C" void kernel_launch(void* const* d_in, const int* in_sizes, int n_in,
                              void* d_out, int out_size, void* d_ws, size_t ws_size,
                              hipStream_t stream) {
    const float*         emb   = (const float*)d_in[0];
    const unsigned char* adj   = (const unsigned char*)d_in[1];
    const float*         W_qk  = (const float*)d_in[2];
    const float*         w_att = (const float*)d_in[3];
    const float*         W_out = (const float*)d_in[4];
    float* out = (float*)d_out;

    float* qbuf = (float*)d_ws;                       // B*H*L*A = 524288 f32
    float* kbuf = qbuf + (size_t)B_ * H_ * L_ * A_;
    float* vals = kbuf + (size_t)B_ * H_ * L_ * A_;

    // K1: 128 x 8 blocks (16x64 tile per wave).
    qk_proj_kernel<<<dim3((B_ * L_) / 16, TWOE_ / 64), 32, 0, stream>>>(
        emb, W_qk, qbuf, kbuf);

    // K2: 16 i-tiles x 64 heads, 256 threads (8 waves) each.
    attn_kernel<<<dim3(L_ / 16, B_ * H_), 256, 0, stream>>>(
        qbuf, kbuf, adj, w_att, vals);

    // K3: 128 x 4 blocks (16x64 tile per wave).
    out_proj_kernel<<<dim3((B_ * L_) / 16, E_ / 64), 32, 0, stream>>>(
        vals, W_out, out);
}